// CompressionModel_77068893160295
// MI455X (gfx1250) — compile-verified
//
#include <hip/hip_runtime.h>

typedef __attribute__((ext_vector_type(2))) float v2f;
typedef __attribute__((ext_vector_type(8))) float v8f;

// ---------------- problem constants ----------------
#define VROWS   66          // embed rows (VOCAB+2)
#define HID     64
#define MAXPOS  64

// ws layout (float offsets). Tables are dense [66][64].
#define WS_TA   0
#define WS_T0   (VROWS * HID)
#define WS_T1   (2 * VROWS * HID)
#define WS_T2   (3 * VROWS * HID)
#define WS_W2P  (4 * VROWS * HID)        // 2048 float2 (pair-packed W2^T)
#define WS_FLOATS (4 * VROWS * HID + 4096)

struct PosCfg {
    int n0, n1, n2;       // positions per level (level0, level1, level2)
    int seqLen;
    float inv_nmem;
    int pos[MAXPOS];      // level0 positions, then level1, then level2
};

// =====================================================================
// Prep: build lookup tables.  For vocab row v:
//   w  = W_f2h @ e          (32)
//   u1 = W_h2f @ w          (64)      level-1 composite * e
//   u2 = W_q2f @ (W_h2q @ w)(64)      level-2 composite * e
//   TA[v][i] = dot(e, W1a[i]) + b1[i]
//   T0[v][i] = dot(e,  W1b[i]) * inv
//   T1[v][i] = dot(u1, W1b[i]) * inv
//   T2[v][i] = dot(u2, W1b[i]) * inv
// Block 0 additionally pair-packs W2 into W2P[kp][v] = (W2[v][2kp], W2[v][2kp+1]).
// =====================================================================
__global__ void cm_prep_tables(const float* __restrict__ embed,
                               const float* __restrict__ W_f2h,
                               const float* __restrict__ W_h2q,
                               const float* __restrict__ W_q2f,
                               const float* __restrict__ W_h2f,
                               const float* __restrict__ W1,
                               const float* __restrict__ b1,
                               const float* __restrict__ W2,
                               float* __restrict__ ws,
                               float inv_nmem)
{
    __shared__ float e[64], w[32], tq[16], u1[64], u2[64];
    const int v = blockIdx.x;
    const int t = threadIdx.x;

    if (t < 64) e[t] = embed[v * 64 + t];
    __syncthreads();

    if (t < 32) {
        float s = 0.f;
        for (int k = 0; k < 64; ++k) s += W_f2h[t * 64 + k] * e[k];
        w[t] = s;
    }
    __syncthreads();

    if (t < 64) {
        float s = 0.f;
        for (int m = 0; m < 32; ++m) s += W_h2f[t * 32 + m] * w[m];
        u1[t] = s;
    } else if (t < 80) {
        const int q = t - 64;
        float s = 0.f;
        for (int m = 0; m < 32; ++m) s += W_h2q[q * 32 + m] * w[m];
        tq[q] = s;
    }
    __syncthreads();

    if (t < 64) {
        float s = 0.f;
        for (int q = 0; q < 16; ++q) s += W_q2f[t * 16 + q] * tq[q];
        u2[t] = s;
    }
    __syncthreads();

    const int tbl = t >> 6;
    const int i   = t & 63;
    float s = 0.f;
    if (tbl == 0) {
        for (int k = 0; k < 64; ++k) s += e[k] * W1[i * 128 + k];
        ws[WS_TA + v * 64 + i] = s + b1[i];
    } else if (tbl == 1) {
        for (int k = 0; k < 64; ++k) s += e[k] * W1[i * 128 + 64 + k];
        ws[WS_T0 + v * 64 + i] = s * inv_nmem;
    } else if (tbl == 2) {
        for (int j = 0; j < 64; ++j) s += u1[j] * W1[i * 128 + 64 + j];
        ws[WS_T1 + v * 64 + i] = s * inv_nmem;
    } else {
        for (int j = 0; j < 64; ++j) s += u2[j] * W1[i * 128 + 64 + j];
        ws[WS_T2 + v * 64 + i] = s * inv_nmem;
    }

    if (v == 0) {
        for (int idx = t; idx < 2048; idx += blockDim.x) {
            const int kp = idx >> 6, vv = idx & 63;
            ws[WS_W2P + 2 * idx    ] = W2[vv * 64 + 2 * kp    ];
            ws[WS_W2P + 2 * idx + 1] = W2[vv * 64 + 2 * kp + 1];
        }
    }
}

// =====================================================================
// Main: one wave handles a tile of 16 batch elements.
//   H[M][k] gathered into v2f h[16] in exactly the WMMA f32 16x16x4
//   A-operand layout (lane M = lane&15, K-pair selected by lane>>4).
//   OUT(16x64) = relu(H) x W2^T + b2 via 16 K-steps x 4 N-tiles of
//   V_WMMA_F32_16X16X4_F32.
// LDS: T1s/T2s/W2Ps with +2*tok (+2*kp) rotation swizzle -> no bank conflicts.
// =====================================================================
__global__ void __launch_bounds__(256)
cm_forward(const int* __restrict__ seqs,
           const int* __restrict__ query_tok,
           const float* __restrict__ b2,
           const float* __restrict__ ws,
           float* __restrict__ out,
           int Btot, PosCfg cfg)
{
    __shared__ float T1s[4096];
    __shared__ float T2s[4096];
    __shared__ v2f   W2Ps[2048];

    const int tid = threadIdx.x;

    // Stage hot tables into LDS with rotation swizzle (row stride 64 would put
    // every lane of a gather on the same bank; rotate columns by 2*tok).
    for (int idx = tid; idx < 4096; idx += blockDim.x) {
        const int tok = idx >> 6, k = idx & 63;
        const int k2 = (k + 2 * tok) & 63;
        T1s[tok * 64 + k2] = ws[WS_T1 + idx];
        T2s[tok * 64 + k2] = ws[WS_T2 + idx];
    }
    for (int idx = tid; idx < 2048; idx += blockDim.x) {
        const int kp = idx >> 6, vv = idx & 63;
        const int v2 = (vv + 2 * kp) & 63;
        W2Ps[kp * 64 + v2] = ((const v2f*)(ws + WS_W2P))[idx];
    }
    __syncthreads();

    const int lane = tid & 31;
    const int wave = tid >> 5;
    const int N  = lane & 15;     // element within tile (gather) / column (B,C)
    const int kh = lane >> 4;     // K-half selector of the WMMA operand layout

    const int ntiles = Btot >> 4;
    const int gwave  = blockIdx.x * (blockDim.x >> 5) + wave;
    const int nwaves = gridDim.x * (blockDim.x >> 5);

    float b2r[4];
#pragma unroll
    for (int nt = 0; nt < 4; ++nt) b2r[nt] = b2[nt * 16 + N];

    for (int tile = gwave; tile < ntiles; tile += nwaves) {
        const int base = tile << 4;
        const int elem = base + N;
        const int srow = elem * cfg.seqLen;

        v2f h[16];
        // query path: TA row (already includes b1), from L2-resident ws
        {
            const int q = query_tok[elem];
            const float* r = ws + WS_TA + q * 64 + 2 * kh;
#pragma unroll
            for (int ks = 0; ks < 16; ++ks) h[ks] = *(const v2f*)(r + 4 * ks);
        }
        // level-0 rows (global; empty in this configuration)
        for (int j = 0; j < cfg.n0; ++j) {
            const int tok = seqs[srow + cfg.pos[j]];
            const float* r = ws + WS_T0 + tok * 64 + 2 * kh;
#pragma unroll
            for (int ks = 0; ks < 16; ++ks) { v2f x = *(const v2f*)(r + 4 * ks); h[ks] += x; }
        }
        // level-1 rows (LDS, swizzled)
        for (int j = 0; j < cfg.n1; ++j) {
            const int tok = seqs[srow + cfg.pos[cfg.n0 + j]];
            const float* r = T1s + tok * 64;
            const int rot = 2 * tok + 2 * kh;
#pragma unroll
            for (int ks = 0; ks < 16; ++ks) {
                v2f x = *(const v2f*)(r + ((4 * ks + rot) & 63));
                h[ks] += x;
            }
        }
        // level-2 rows (LDS, swizzled)
        for (int j = 0; j < cfg.n2; ++j) {
            const int tok = seqs[srow + cfg.pos[cfg.n0 + cfg.n1 + j]];
            const float* r = T2s + tok * 64;
            const int rot = 2 * tok + 2 * kh;
#pragma unroll
            for (int ks = 0; ks < 16; ++ks) {
                v2f x = *(const v2f*)(r + ((4 * ks + rot) & 63));
                h[ks] += x;
            }
        }
        // ReLU
#pragma unroll
        for (int ks = 0; ks < 16; ++ks) {
            h[ks].x = fmaxf(h[ks].x, 0.f);
            h[ks].y = fmaxf(h[ks].y, 0.f);
        }

        // OUT = relu(H) @ W2^T + b2 : four independent WMMA accumulation chains
        v8f c0, c1, c2, c3;
#pragma unroll
        for (int r = 0; r < 8; ++r) { c0[r] = b2r[0]; c1[r] = b2r[1]; c2[r] = b2r[2]; c3[r] = b2r[3]; }

#pragma unroll
        for (int ks = 0; ks < 16; ++ks) {
            const int kp = 2 * ks + kh;
            const v2f* bp = W2Ps + kp * 64;
            const int r2 = 2 * kp;
            const v2f bv0 = bp[(0 * 16 + N + r2) & 63];
            const v2f bv1 = bp[(1 * 16 + N + r2) & 63];
            const v2f bv2 = bp[(2 * 16 + N + r2) & 63];
            const v2f bv3 = bp[(3 * 16 + N + r2) & 63];
            c0 = __builtin_amdgcn_wmma_f32_16x16x4_f32(false, h[ks], false, bv0, (short)0, c0, false, false);
            c1 = __builtin_amdgcn_wmma_f32_16x16x4_f32(false, h[ks], false, bv1, (short)0, c1, false, false);
            c2 = __builtin_amdgcn_wmma_f32_16x16x4_f32(false, h[ks], false, bv2, (short)0, c2, false, false);
            c3 = __builtin_amdgcn_wmma_f32_16x16x4_f32(false, h[ks], false, bv3, (short)0, c3, false, false);
        }

        // store D tiles: VGPR r -> row (base + r + 8*kh), column N of each N-tile
#pragma unroll
        for (int r = 0; r < 8; ++r) {
            const int row = base + r + 8 * kh;
            float* o = out + row * 64 + N;
            o[0]  = c0[r];
            o[16] = c1[r];
            o[32] = c2[r];
            o[48] = c3[r];
        }
    }
}

// =====================================================================
// Host: replay the compress/evict loop (data-independent), launch.
// =====================================================================
static PosCfg cm_make_cfg(int seqLen)
{
    const int LD[3] = {64, 32, 16};
    int pos_[MAXPOS], lvl_[MAXPOS];
    int n = 0;
    auto total = [&]() { long s = 0; for (int i = 0; i < n; ++i) s += LD[lvl_[i]]; return s; };
    for (int t = 0; t < seqLen - 1; ++t) {
        pos_[n] = t; lvl_[n] = 0; ++n;
        while (total() > 256) {
            bool ok = false;
            for (int i = 0; i < n; ++i) {
                if (lvl_[i] < 2) {
                    lvl_[i]++;
                    if (total() <= 256) { ok = true; break; }
                }
            }
            if (!ok && n > 0) {
                for (int i = 1; i < n; ++i) { pos_[i-1] = pos_[i]; lvl_[i-1] = lvl_[i]; }
                --n;
            }
        }
    }
    PosCfg c{};
    c.seqLen = seqLen;
    int idx = 0;
    for (int l = 0; l < 3; ++l)
        for (int i = 0; i < n; ++i)
            if (lvl_[i] == l) c.pos[idx++] = pos_[i];
    c.n0 = c.n1 = c.n2 = 0;
    for (int i = 0; i < n; ++i) {
        if (lvl_[i] == 0) c.n0++;
        else if (lvl_[i] == 1) c.n1++;
        else c.n2++;
    }
    c.inv_nmem = 1.0f / (float)(n > 0 ? n : 1);
    return c;
}

extern "C" void kernel_launch(void* const* d_in, const int* in_sizes, int n_in,
                              void* d_out, int out_size, void* d_ws, size_t ws_size,
                              hipStream_t stream)
{
    const int*   seqs   = (const int*)  d_in[0];
    const int*   query  = (const int*)  d_in[1];
    const float* embed  = (const float*)d_in[2];
    const float* W_f2h  = (const float*)d_in[3];
    const float* W_h2q  = (const float*)d_in[4];
    const float* W_q2f  = (const float*)d_in[5];
    const float* W_h2f  = (const float*)d_in[6];
    const float* W1     = (const float*)d_in[7];
    const float* b1     = (const float*)d_in[8];
    const float* W2     = (const float*)d_in[9];
    const float* b2     = (const float*)d_in[10];
    float* out = (float*)d_out;
    float* ws  = (float*)d_ws;

    const int B      = in_sizes[1];
    const int seqLen = in_sizes[0] / (B > 0 ? B : 1);

    PosCfg cfg = cm_make_cfg(seqLen);

    cm_prep_tables<<<VROWS, 256, 0, stream>>>(embed, W_f2h, W_h2q, W_q2f, W_h2f,
                                              W1, b1, W2, ws, cfg.inv_nmem);

    const int ntiles = B >> 4;                       // 16 elements per wave-tile
    int blocks = (ntiles + 8 * 4 - 1) / (8 * 4);     // 8 waves/block, ~4 tiles/wave
    if (blocks < 1) blocks = 1;
    if (blocks > 2048) blocks = 2048;
    cm_forward<<<blocks, 256, 0, stream>>>(seqs, query, b2, ws, out, B, cfg);
}